// GCN_83872121356313
// MI455X (gfx1250) — compile-verified
//
#include <hip/hip_runtime.h>

typedef __attribute__((ext_vector_type(2))) float v2f;
typedef __attribute__((ext_vector_type(8))) float v8f;

// ---------------------------------------------------------------------------
// Utility: zero a float buffer (grid-stride).
// ---------------------------------------------------------------------------
__global__ void zero_f32(float* __restrict__ p, size_t n) {
  size_t i = (size_t)blockIdx.x * blockDim.x + threadIdx.x;
  size_t stride = (size_t)gridDim.x * blockDim.x;
  for (; i < n; i += stride) p[i] = 0.0f;
}

// ---------------------------------------------------------------------------
// Degree: deg[dst[e]] += 1  (hardware f32 atomic, L2-resident)
// ---------------------------------------------------------------------------
__global__ void degree_kernel(const long long* __restrict__ dst,
                              float* __restrict__ deg, int E) {
  int e = blockIdx.x * blockDim.x + threadIdx.x;
  if (e < E) unsafeAtomicAdd(&deg[(int)dst[e]], 1.0f);
}

// dinv = rsqrt(deg + 1)   (in place)
__global__ void dinv_kernel(float* __restrict__ deg, int n) {
  int i = blockIdx.x * blockDim.x + threadIdx.x;
  if (i < n) deg[i] = rsqrtf(deg[i] + 1.0f);
}

// ---------------------------------------------------------------------------
// GEMM: H[M x N] = X[M x K] * W[K x N], fp32 WMMA 16x16x4.
// One wave owns a 16-row strip and the full N extent (N/16 accumulators),
// reusing the A fragment across all N tiles per k-step.
//
// Fragment layouts (wave32, CDNA5 ISA):
//   A 16x4 : lanes 0-15 -> M=lane, VGPR0/1 = K 0/1 ; lanes 16-31 -> K 2/3
//   B 4x16 : lanes 0-15 -> N=lane, VGPR0/1 = K 0/1 ; lanes 16-31 -> K 2/3
//   C 16x16: VGPR r, lanes 0-15 -> M=r, N=lane ; lanes 16-31 -> M=r+8
// ---------------------------------------------------------------------------
template <int K_DIM, int N_DIM>
__global__ void gemm_wmma_f32(const float* __restrict__ X,
                              const float* __restrict__ W,
                              float* __restrict__ H, int M) {
  const int wave = threadIdx.x >> 5;
  const int lane = threadIdx.x & 31;
  const int mt   = blockIdx.x * (blockDim.x >> 5) + wave;
  if (mt * 16 >= M) return;                 // wave-uniform: EXEC stays all-1s

  const int half = lane >> 4;               // 0: K 0/1, 1: K 2/3
  const int lr   = lane & 15;

  constexpr int NT = N_DIM / 16;
  v8f acc[NT] = {};

  const float* xrow = X + (size_t)(mt * 16 + lr) * K_DIM;

  for (int k = 0; k < K_DIM; k += 4) {
    // A fragment: two consecutive K values for row `lr` (8B-aligned load)
    const v2f a = *(const v2f*)(xrow + k + 2 * half);
#pragma unroll
    for (int nt = 0; nt < NT; nt++) {
      v2f b;
      b.x = W[(size_t)(k + 2 * half + 0) * N_DIM + nt * 16 + lr];
      b.y = W[(size_t)(k + 2 * half + 1) * N_DIM + nt * 16 + lr];
      acc[nt] = __builtin_amdgcn_wmma_f32_16x16x4_f32(
          /*neg_a=*/false, a, /*neg_b=*/false, b,
          /*c_mod=*/(short)0, acc[nt], /*reuse_a=*/false, /*reuse_b=*/false);
    }
  }

#pragma unroll
  for (int nt = 0; nt < NT; nt++) {
#pragma unroll
    for (int r = 0; r < 8; r++) {
      const int row = mt * 16 + r + 8 * half;
      H[(size_t)row * N_DIM + nt * 16 + lr] = acc[nt][r];
    }
  }
}

// ---------------------------------------------------------------------------
// Edge scatter: agg[dst] += h[src] * (dinv[src]*dinv[dst])
// DIM/4 threads per edge, float4 per thread, 4 hardware f32 atomics.
// agg (25.6 MB max) lives in L2, so atomics resolve at L2 bandwidth.
// ---------------------------------------------------------------------------
template <int DIM>
__global__ void scatter_edges(const long long* __restrict__ src,
                              const long long* __restrict__ dst,
                              const float* __restrict__ dinv,
                              const float* __restrict__ H,
                              float* __restrict__ AGG, int E) {
  constexpr int CH = DIM / 4;
  int idx = blockIdx.x * blockDim.x + threadIdx.x;
  int e = idx / CH;
  if (e >= E) return;
  int c = (idx % CH) * 4;

  const int s = (int)src[e];
  const int d = (int)dst[e];
  const float norm = dinv[s] * dinv[d];

  const float4 v = *(const float4*)(H + (size_t)s * DIM + c);
  float* p = AGG + (size_t)d * DIM + c;
  unsafeAtomicAdd(p + 0, v.x * norm);
  unsafeAtomicAdd(p + 1, v.y * norm);
  unsafeAtomicAdd(p + 2, v.z * norm);
  unsafeAtomicAdd(p + 3, v.w * norm);
}

// ---------------------------------------------------------------------------
// Combine: out = agg + h * dinv^2 + b   (+ ReLU), float4 vectorized.
// ---------------------------------------------------------------------------
template <int DIM, bool RELU>
__global__ void combine_kernel(const float* __restrict__ H,
                               const float* __restrict__ AGG,
                               const float* __restrict__ dinv,
                               const float* __restrict__ bias,
                               float* __restrict__ OUT, int M) {
  constexpr int CH = DIM / 4;
  int idx = blockIdx.x * blockDim.x + threadIdx.x;
  if (idx >= M * CH) return;
  int i = idx / CH;
  int c = (idx % CH) * 4;

  const float di = dinv[i];
  const float w  = di * di;
  const float4 h = *(const float4*)(H + (size_t)i * DIM + c);
  const float4 a = *(const float4*)(AGG + (size_t)i * DIM + c);
  const float4 b = *(const float4*)(bias + c);

  float4 o;
  o.x = a.x + h.x * w + b.x;
  o.y = a.y + h.y * w + b.y;
  o.z = a.z + h.z * w + b.z;
  o.w = a.w + h.w * w + b.w;
  if (RELU) {
    o.x = fmaxf(o.x, 0.0f);
    o.y = fmaxf(o.y, 0.0f);
    o.z = fmaxf(o.z, 0.0f);
    o.w = fmaxf(o.w, 0.0f);
  }
  *(float4*)(OUT + (size_t)i * DIM + c) = o;
}

// ---------------------------------------------------------------------------
// Host launcher
// Inputs (setup_inputs order): x, edge_index, W0, b0, W1, b1, W2, b2
// ---------------------------------------------------------------------------
extern "C" void kernel_launch(void* const* d_in, const int* in_sizes, int n_in,
                              void* d_out, int out_size, void* d_ws, size_t ws_size,
                              hipStream_t stream) {
  const float*     x  = (const float*)d_in[0];
  const long long* ei = (const long long*)d_in[1];   // int64, shape (2, E)
  const float*     W0 = (const float*)d_in[2];
  const float*     b0 = (const float*)d_in[3];
  const float*     W1 = (const float*)d_in[4];
  const float*     b1 = (const float*)d_in[5];
  const float*     W2 = (const float*)d_in[6];
  const float*     b2 = (const float*)d_in[7];

  const int E  = in_sizes[1] / 2;
  const int Nn = in_sizes[0] / 256;                  // 50000 nodes

  const long long* src = ei;
  const long long* dst = ei + E;

  // Workspace layout: deg/dinv | H (gemm out) | AGG | ACT (layer activations)
  float* deg = (float*)d_ws;                         // Nn
  float* H   = deg + Nn;                             // Nn*128
  float* AGG = H + (size_t)Nn * 128;                 // Nn*128
  float* ACT = AGG + (size_t)Nn * 128;               // Nn*128

  const int B = 256;

  // ---- degrees -> dinv (in place) ----
  zero_f32<<<1024, B, 0, stream>>>(deg, (size_t)Nn);
  degree_kernel<<<(E + B - 1) / B, B, 0, stream>>>(dst, deg, E);
  dinv_kernel<<<(Nn + B - 1) / B, B, 0, stream>>>(deg, Nn);

  const int mtiles = (Nn + 15) / 16;                 // 3125 (exact)
  const int gemm_blocks = (mtiles + 7) / 8;          // 8 waves / block

  // ---- layer 0: 256 -> 128, ReLU ----
  gemm_wmma_f32<256, 128><<<gemm_blocks, B, 0, stream>>>(x, W0, H, Nn);
  zero_f32<<<4096, B, 0, stream>>>(AGG, (size_t)Nn * 128);
  {
    const long long th = (long long)E * 32;
    scatter_edges<128><<<(int)((th + B - 1) / B), B, 0, stream>>>(src, dst, deg, H, AGG, E);
  }
  combine_kernel<128, true><<<(Nn * 32 + B - 1) / B, B, 0, stream>>>(H, AGG, deg, b0, ACT, Nn);

  // ---- layer 1: 128 -> 128, ReLU ----
  gemm_wmma_f32<128, 128><<<gemm_blocks, B, 0, stream>>>(ACT, W1, H, Nn);
  zero_f32<<<4096, B, 0, stream>>>(AGG, (size_t)Nn * 128);
  {
    const long long th = (long long)E * 32;
    scatter_edges<128><<<(int)((th + B - 1) / B), B, 0, stream>>>(src, dst, deg, H, AGG, E);
  }
  combine_kernel<128, true><<<(Nn * 32 + B - 1) / B, B, 0, stream>>>(H, AGG, deg, b1, ACT, Nn);

  // ---- layer 2: 128 -> 64, no ReLU, write d_out ----
  gemm_wmma_f32<128, 64><<<gemm_blocks, B, 0, stream>>>(ACT, W2, H, Nn);
  zero_f32<<<4096, B, 0, stream>>>(AGG, (size_t)Nn * 64);
  {
    const long long th = (long long)E * 16;
    scatter_edges<64><<<(int)((th + B - 1) / B), B, 0, stream>>>(src, dst, deg, H, AGG, E);
  }
  combine_kernel<64, false><<<(Nn * 16 + B - 1) / B, B, 0, stream>>>(H, AGG, deg, b2, (float*)d_out, Nn);
}